// QuantumAttentionCore_83863531421735
// MI455X (gfx1250) — compile-verified
//
#include <hip/hip_runtime.h>

// ---- problem constants (match reference) ----
#define NQ    4
#define DIM   16          // 2^NQ
#define TSTEP 200         // T
#define BATCH 2048        // B
#define NDEG  3           // qsvt degree
#define NROT1 20          // rotations per ansatz layer (5*NQ)

typedef float v2f __attribute__((ext_vector_type(2)));
typedef float v8f __attribute__((ext_vector_type(8)));

// ---------------- gate primitives on a 16-amplitude register state ----------------
// state: float2 s[16], s[j].x = Re, s[j].y = Im. Qubit w pairs indices differing
// in bit (8 >> w) (PennyLane wire-0-most-significant ordering).

__device__ __forceinline__ void g_rx(float2* s, int bit, float c, float sn) {
#pragma unroll
  for (int i0 = 0; i0 < DIM; ++i0) {
    if (i0 & bit) continue;
    const int i1 = i0 | bit;
    const float2 a = s[i0], b = s[i1];
    // [[c, -i s],[-i s, c]]
    s[i0].x = fmaf(c, a.x,  sn * b.y);
    s[i0].y = fmaf(c, a.y, -sn * b.x);
    s[i1].x = fmaf(c, b.x,  sn * a.y);
    s[i1].y = fmaf(c, b.y, -sn * a.x);
  }
}

__device__ __forceinline__ void g_ry(float2* s, int bit, float c, float sn) {
#pragma unroll
  for (int i0 = 0; i0 < DIM; ++i0) {
    if (i0 & bit) continue;
    const int i1 = i0 | bit;
    const float2 a = s[i0], b = s[i1];
    // [[c, -s],[s, c]] (real)
    s[i0].x = fmaf(c, a.x, -sn * b.x);
    s[i0].y = fmaf(c, a.y, -sn * b.y);
    s[i1].x = fmaf(c, b.x,  sn * a.x);
    s[i1].y = fmaf(c, b.y,  sn * a.y);
  }
}

__device__ __forceinline__ void g_rz(float2* s, int bit, float c, float sn) {
#pragma unroll
  for (int i0 = 0; i0 < DIM; ++i0) {
    if (i0 & bit) continue;
    const int i1 = i0 | bit;
    const float2 a = s[i0], b = s[i1];
    // diag(e^{-ih}, e^{+ih})
    s[i0].x = fmaf(c, a.x,  sn * a.y);
    s[i0].y = fmaf(c, a.y, -sn * a.x);
    s[i1].x = fmaf(c, b.x, -sn * b.y);
    s[i1].y = fmaf(c, b.y,  sn * b.x);
  }
}

__device__ __forceinline__ void g_crx(float2* s, int cbit, int tbit, float c, float sn) {
#pragma unroll
  for (int i0 = 0; i0 < DIM; ++i0) {
    if (!(i0 & cbit) || (i0 & tbit)) continue;   // control=1, target=0
    const int i1 = i0 | tbit;
    const float2 a = s[i0], b = s[i1];
    s[i0].x = fmaf(c, a.x,  sn * b.y);
    s[i0].y = fmaf(c, a.y, -sn * b.x);
    s[i1].x = fmaf(c, b.x,  sn * a.y);
    s[i1].y = fmaf(c, b.y, -sn * a.x);
  }
}

// Full ansatz; h[] holds half-angles (param * 0.5), 20 per layer.
template <int LAYERS>
__device__ __forceinline__ void ansatz(float2* s, const float* h) {
  int idx = 0;
#pragma unroll
  for (int L = 0; L < LAYERS; ++L) {
#pragma unroll
    for (int i = 0; i < NQ; ++i) {
      const int bit = 8 >> i;
      float sn, c;
      __sincosf(h[idx + 0], &sn, &c); g_rx(s, bit, c, sn);
      __sincosf(h[idx + 1], &sn, &c); g_ry(s, bit, c, sn);
      __sincosf(h[idx + 2], &sn, &c); g_rz(s, bit, c, sn);
      idx += 3;
    }
#pragma unroll
    for (int i = 0; i < NQ; ++i) {            // control i -> target (i+1)%4
      float sn, c; __sincosf(h[idx], &sn, &c);
      g_crx(s, 8 >> i, 8 >> ((i + 1) & 3), c, sn);
      ++idx;
    }
#pragma unroll
    for (int i = NQ - 1; i >= 0; --i) {       // control i -> target (i-1)%4
      float sn, c; __sincosf(h[idx], &sn, &c);
      g_crx(s, 8 >> i, 8 >> ((i + 3) & 3), c, sn);
      ++idx;
    }
  }
}

// ---------------- kernels ----------------

// ws layout (floats): work[B*32] | acc[B*32]; per state: [2j]=Re(amp j), [2j+1]=Im.
__global__ void qac_init(float* __restrict__ work, float* __restrict__ accb,
                         const float* __restrict__ poly) {
  const int i = blockIdx.x * blockDim.x + threadIdx.x;
  if (i < BATCH * 32) {
    const bool z = (i & 31) == 0;
    work[i] = z ? 1.0f : 0.0f;
    accb[i] = z ? poly[0] : 0.0f;
  }
}

// One block per batch element. Threads 0..199 evolve the state for their t;
// wave 0 then does the K=2T contraction with v_wmma_f32_16x16x4_f32.
__global__ __launch_bounds__(256) void qac_lcu_step(
    float* __restrict__ work, float* __restrict__ accb,
    const float* __restrict__ tparams,
    const float* __restrict__ mre, const float* __restrict__ mim,
    const float* __restrict__ poly, int k)
{
  __shared__ float s_in[32];
  __shared__ float s_u[2 * TSTEP];        // weights for Re output column (N=0)
  __shared__ float s_v[2 * TSTEP];        // weights for Im output column (N=1)
  __shared__ float s_ev[TSTEP * 32];      // ev[kappa=2t+p][amp m] at 16*kappa + m

  const int b   = blockIdx.x;
  const int tid = threadIdx.x;

  if (tid < 32) s_in[tid] = work[b * 32 + tid];
  if (tid < TSTEP) {
    const float r = mre[tid], m = mim[tid];
    s_u[2 * tid]     = r;   // Re: +evre*mre
    s_u[2 * tid + 1] = -m;  //     -evim*mim
    s_v[2 * tid]     = m;   // Im: +evre*mim
    s_v[2 * tid + 1] = r;   //     +evim*mre
  }
  __syncthreads();

  if (tid < TSTEP) {
    float2 st[DIM];
#pragma unroll
    for (int j = 0; j < DIM; ++j) { st[j].x = s_in[2 * j]; st[j].y = s_in[2 * j + 1]; }

    // 40 params per (b,t); rows are 160B (16B aligned) -> 10x global_load_b128
    float h[2 * NROT1];
    const float4* p4 = (const float4*)(tparams + ((size_t)b * TSTEP + tid) * (2 * NROT1));
#pragma unroll
    for (int r = 0; r < NROT1 / 2; ++r) {
      const float4 v = p4[r];
      h[4 * r + 0] = 0.5f * v.x;
      h[4 * r + 1] = 0.5f * v.y;
      h[4 * r + 2] = 0.5f * v.z;
      h[4 * r + 3] = 0.5f * v.w;
    }

    ansatz<2>(st, h);

#pragma unroll
    for (int j = 0; j < DIM; ++j) {
      s_ev[32 * tid + j]      = st[j].x;   // kappa = 2t   (real part)
      s_ev[32 * tid + 16 + j] = st[j].y;   // kappa = 2t+1 (imag part)
    }
  }
  __syncthreads();

  if (tid < 32) {                 // wave 0, EXEC all ones -> WMMA legal
    const int half = tid >> 4;    // K half-select (lanes 16-31 -> K=2,3)
    const int m    = tid & 15;    // A: M row / B,D: N column
    // Branch-free B-matrix: per-lane weight pointer + 0/1 mask (no exec games).
    const float* wsel = (m == 1) ? s_v : s_u;
    const float  msk  = (m < 2) ? 1.0f : 0.0f;

    // Two independent accumulation chains over K = 2T = 400 (2 x 50 chunks of 4).
    v8f acc0 = {}, acc1 = {};
#pragma unroll 2
    for (int chunk = 0; chunk < (2 * TSTEP) / 8; ++chunk) {
      const int ka = 4 * chunk + 2 * half;            // kappa 0..199
      const int kb = ka + (2 * TSTEP) / 2;            // kappa 200..399
      v2f a0, b0, a1, b1;
      a0.x = s_ev[16 * ka + m];
      a0.y = s_ev[16 * ka + 16 + m];
      b0.x = msk * wsel[ka];
      b0.y = msk * wsel[ka + 1];
      a1.x = s_ev[16 * kb + m];
      a1.y = s_ev[16 * kb + 16 + m];
      b1.x = msk * wsel[kb];
      b1.y = msk * wsel[kb + 1];
      acc0 = __builtin_amdgcn_wmma_f32_16x16x4_f32(false, a0, false, b0,
                                                   (short)0, acc0, false, false);
      acc1 = __builtin_amdgcn_wmma_f32_16x16x4_f32(false, a1, false, b1,
                                                   (short)0, acc1, false, false);
    }
    // D[M][N]: lane (N + 16*hs), VGPR r -> M = 8*hs + r. Column 0 = Re, 1 = Im.
    if (m < 2) {
      const float pk = poly[k];
#pragma unroll
      for (int r = 0; r < 8; ++r) {
        const int amp = 8 * half + r;
        const int j   = 2 * amp + m;        // interleaved re/im slot
        const float val = acc0[r] + acc1[r];
        work[b * 32 + j] = val;
        accb[b * 32 + j] += pk * val;
      }
    }
  }
}

// Epilogue: L1/L2 normalize, 1-layer qff ansatz, expectation values. 1 thread / b.
__global__ __launch_bounds__(256) void qac_final(
    const float* __restrict__ accb, const float* __restrict__ poly,
    const float* __restrict__ qff, float* __restrict__ out)
{
  const int b = blockIdx.x * blockDim.x + threadIdx.x;
  if (b >= BATCH) return;

  const float l1 = fabsf(poly[0]) + fabsf(poly[1]) + fabsf(poly[2]) + fabsf(poly[3]);
  const float il1 = 1.0f / l1;

  float2 st[DIM];
  float n2 = 0.0f;
#pragma unroll
  for (int j = 0; j < DIM; ++j) {
    const float re = accb[b * 32 + 2 * j] * il1;
    const float im = accb[b * 32 + 2 * j + 1] * il1;
    st[j].x = re; st[j].y = im;
    n2 = fmaf(re, re, fmaf(im, im, n2));
  }
  const float inv = 1.0f / (sqrtf(n2) + 1e-9f);
#pragma unroll
  for (int j = 0; j < DIM; ++j) { st[j].x *= inv; st[j].y *= inv; }

  float h[NROT1];
#pragma unroll
  for (int r = 0; r < NROT1; ++r) h[r] = 0.5f * qff[r];
  ansatz<1>(st, h);

#pragma unroll
  for (int i = 0; i < NQ; ++i) {
    const int bit = 8 >> i;
    float X = 0.0f, Y = 0.0f, Z = 0.0f;
#pragma unroll
    for (int i0 = 0; i0 < DIM; ++i0) {
      if (i0 & bit) continue;
      const float2 a = st[i0], c = st[i0 | bit];
      X += 2.0f * (a.x * c.x + a.y * c.y);       // 2 Re(conj(s0)*s1)
      Y += 2.0f * (a.x * c.y - a.y * c.x);       // 2 Im(conj(s0)*s1)
      Z += (a.x * a.x + a.y * a.y) - (c.x * c.x + c.y * c.y);
    }
    out[b * 12 + i]     = X;
    out[b * 12 + 4 + i] = Y;
    out[b * 12 + 8 + i] = Z;
  }
}

// ---------------- host launch ----------------
extern "C" void kernel_launch(void* const* d_in, const int* in_sizes, int n_in,
                              void* d_out, int out_size, void* d_ws, size_t ws_size,
                              hipStream_t stream) {
  const float* tparams = (const float*)d_in[0];  // (B, T, 40)
  const float* poly    = (const float*)d_in[1];  // (4,)
  const float* mre     = (const float*)d_in[2];  // (T,)
  const float* mim     = (const float*)d_in[3];  // (T,)
  const float* qff     = (const float*)d_in[4];  // (20,)
  float* out = (float*)d_out;                    // (B, 12)

  float* ws   = (float*)d_ws;
  float* work = ws;                 // B*32 floats
  float* accb = ws + BATCH * 32;    // B*32 floats

  qac_init<<<(BATCH * 32 + 255) / 256, 256, 0, stream>>>(work, accb, poly);

  for (int k = 1; k <= NDEG; ++k)
    qac_lcu_step<<<BATCH, 256, 0, stream>>>(work, accb, tparams, mre, mim, poly, k);

  qac_final<<<BATCH / 256, 256, 0, stream>>>(accb, poly, qff, out);
}